// VIFLoss_88313117540651
// MI455X (gfx1250) — compile-verified
//
#include <hip/hip_runtime.h>
#include <math.h>

typedef __attribute__((ext_vector_type(2))) float v2f;
typedef __attribute__((ext_vector_type(8))) float v8f;

#define IMG_H 135
#define IMG_W 240
#define N_IMG 384              // B*C
#define N_B   128
#define N_C   3
#define CHW   (N_C * IMG_H * IMG_W)
#define EPS_V 1e-10f

// ---------------------------------------------------------------------------
// Per-scale VIF tile kernel. One wave (32 threads) computes one 16x16 output
// tile for one image: 5 separable Gaussian convolutions (r, p, r^2, p^2, r*p)
// done as Toeplitz matmuls on v_wmma_f32_16x16x4_f32, then the VIF statistic
// and log10 reduction, accumulated into per-image num/den via atomics.
// ---------------------------------------------------------------------------
template <int N>
__global__ __launch_bounds__(32)
void vif_scale_kernel(const float* __restrict__ recons,
                      const float* __restrict__ xin,
                      float* __restrict__ numAcc,
                      float* __restrict__ denAcc)
{
    constexpr int R  = 16 + N - 1;        // extended extent (<= 32)
    constexpr int KS = (R + 3) / 4;       // K-slices of 4 (Toeplitz padded w/ 0)
    constexpr int HO = IMG_H - N + 1;
    constexpr int WO = IMG_W - N + 1;

    __shared__ float shR[32][36];         // stride 36: 16B-aligned float4 rows
    __shared__ float shP[32][36];
    __shared__ float shT[5][32][18];      // tmp planes (even stride)
    __shared__ float shW[64];             // zero-padded weights at [32..32+N)

    const int lane = threadIdx.x;   // 0..31
    const int col  = lane & 15;
    const int half = lane >> 4;
    const int img  = blockIdx.z;
    const int x0   = blockIdx.x * 16;
    const int y0   = blockIdx.y * 16;

    const size_t base = (size_t)img * (IMG_H * IMG_W);
    const float* rsrc = recons + base;
    const float* psrc = xin + base;

    // ---- zero-padded, normalized 1-D Gaussian weight table -----------------
    shW[lane] = 0.f;
    shW[32 + lane] = 0.f;
    __syncthreads();
    if (lane < N) {
        const float sigma = (float)N / 5.0f;
        const float d = (float)(lane - (N / 2));
        shW[32 + lane] = __expf(-(d * d) / (2.0f * sigma * sigma));
    }
    __syncthreads();
    if (lane == 0) {
        float s = 0.f;
        for (int j = 0; j < N; ++j) s += shW[32 + j];
        shW[0] = 1.0f / s;
    }
    __syncthreads();
    if (lane < N) shW[32 + lane] *= shW[0];

    // ---- load zero-padded 32x32 tiles of r and p (vectorized) --------------
    for (int i = lane; i < 256; i += 32) {
        const int row = i >> 3;
        const int c4  = (i & 7) << 2;
        const int gy  = y0 + row;
        const int gx  = x0 + c4;
        float4 rv = make_float4(0.f, 0.f, 0.f, 0.f);
        float4 pv = make_float4(0.f, 0.f, 0.f, 0.f);
        if (gy < IMG_H) {
            const int off = gy * IMG_W + gx;
            if (gx + 3 < IMG_W) {            // 16B-aligned fast path
                rv = *(const float4*)(rsrc + off);
                pv = *(const float4*)(psrc + off);
            } else {
                #pragma unroll
                for (int j = 0; j < 4; ++j)
                    if (gx + j < IMG_W) {
                        ((float*)&rv)[j] = rsrc[off + j];
                        ((float*)&pv)[j] = psrc[off + j];
                    }
            }
        }
        *(float4*)&shR[row][c4] = rv;
        *(float4*)&shP[row][c4] = pv;
    }
    __syncthreads();

    // ---- Toeplitz weight fragments (register-resident, shared by both
    //      passes: Th[k][x'] = w[k-x'] and Tv[m][k] = w[k-m], m==col) --------
    v2f wf[KS];
    #pragma unroll
    for (int ks = 0; ks < KS; ++ks) {
        const int kA = 4 * ks + 2 * half;
        wf[ks].x = shW[32 + kA - col];       // always in [17,63] -> no branch
        wf[ks].y = shW[32 + kA + 1 - col];
    }

    // ---- horizontal pass, all 5 planes simultaneously ----------------------
    // Tmp_q(R x 16) = In_q(R x R) * Th(R x 16)
    v8f h[10];
    #pragma unroll
    for (int q = 0; q < 10; ++q) h[q] = (v8f){0.f,0.f,0.f,0.f,0.f,0.f,0.f,0.f};

    #pragma unroll
    for (int ks = 0; ks < KS; ++ks) {
        const int kA = 4 * ks + 2 * half;           // even -> 8B aligned
        const float2 rA0 = *(const float2*)&shR[col][kA];
        const float2 rA1 = *(const float2*)&shR[16 + col][kA];
        const float2 pA0 = *(const float2*)&shP[col][kA];
        const float2 pA1 = *(const float2*)&shP[16 + col][kA];
        const v2f b = wf[ks];
        v2f a;
        // q=0 : r
        a.x = rA0.x;         a.y = rA0.y;
        h[0] = __builtin_amdgcn_wmma_f32_16x16x4_f32(false, a, false, b, (short)0, h[0], false, false);
        a.x = rA1.x;         a.y = rA1.y;
        h[1] = __builtin_amdgcn_wmma_f32_16x16x4_f32(false, a, false, b, (short)0, h[1], false, false);
        // q=1 : p
        a.x = pA0.x;         a.y = pA0.y;
        h[2] = __builtin_amdgcn_wmma_f32_16x16x4_f32(false, a, false, b, (short)0, h[2], false, false);
        a.x = pA1.x;         a.y = pA1.y;
        h[3] = __builtin_amdgcn_wmma_f32_16x16x4_f32(false, a, false, b, (short)0, h[3], false, false);
        // q=2 : r*r
        a.x = rA0.x * rA0.x; a.y = rA0.y * rA0.y;
        h[4] = __builtin_amdgcn_wmma_f32_16x16x4_f32(false, a, false, b, (short)0, h[4], false, false);
        a.x = rA1.x * rA1.x; a.y = rA1.y * rA1.y;
        h[5] = __builtin_amdgcn_wmma_f32_16x16x4_f32(false, a, false, b, (short)0, h[5], false, false);
        // q=3 : p*p
        a.x = pA0.x * pA0.x; a.y = pA0.y * pA0.y;
        h[6] = __builtin_amdgcn_wmma_f32_16x16x4_f32(false, a, false, b, (short)0, h[6], false, false);
        a.x = pA1.x * pA1.x; a.y = pA1.y * pA1.y;
        h[7] = __builtin_amdgcn_wmma_f32_16x16x4_f32(false, a, false, b, (short)0, h[7], false, false);
        // q=4 : r*p
        a.x = rA0.x * pA0.x; a.y = rA0.y * pA0.y;
        h[8] = __builtin_amdgcn_wmma_f32_16x16x4_f32(false, a, false, b, (short)0, h[8], false, false);
        a.x = rA1.x * pA1.x; a.y = rA1.y * pA1.y;
        h[9] = __builtin_amdgcn_wmma_f32_16x16x4_f32(false, a, false, b, (short)0, h[9], false, false);
    }

    // ---- scatter D fragments (documented f32 C/D layout) to tmp planes ----
    __syncthreads();
    #pragma unroll
    for (int q = 0; q < 5; ++q) {
        #pragma unroll
        for (int e = 0; e < 8; ++e) {
            shT[q][e + 8 * half][col]      = h[2 * q][e];
            shT[q][16 + e + 8 * half][col] = h[2 * q + 1][e];
        }
    }
    __syncthreads();

    // ---- vertical pass: Out_q(16x16) = Tv(16 x R) * Tmp_q(R x 16) ----------
    v8f o[5];
    #pragma unroll
    for (int q = 0; q < 5; ++q) o[q] = (v8f){0.f,0.f,0.f,0.f,0.f,0.f,0.f,0.f};

    #pragma unroll
    for (int ks = 0; ks < KS; ++ks) {
        const int kA = 4 * ks + 2 * half;
        const v2f a = wf[ks];                // Tv slice == Th slice (m == col)
        #pragma unroll
        for (int q = 0; q < 5; ++q) {
            v2f b;
            b.x = shT[q][kA][col];
            b.y = shT[q][kA + 1][col];
            o[q] = __builtin_amdgcn_wmma_f32_16x16x4_f32(false, a, false, b, (short)0, o[q], false, false);
        }
    }

    // ---- per-pixel VIF statistics + log10 reduction ------------------------
    // fast-math intrinsics: v_log_f32 / v_rcp_f32 based, ~1 ulp -- more than
    // enough accuracy for a scalar loss averaged over ~12M pixels.
    float nsum = 0.f, dsum = 0.f;
    #pragma unroll
    for (int e = 0; e < 8; ++e) {
        const int Y = y0 + e + 8 * half;
        const int X = x0 + col;
        if (Y < HO && X < WO) {
            const float mu1 = o[0][e], mu2 = o[1][e];
            const float err = o[2][e], epp = o[3][e], erp = o[4][e];
            float sGT = fmaxf(err - mu1 * mu1, 0.f);
            float sP  = fmaxf(epp - mu2 * mu2, 0.f);
            const float sGP = erp - mu1 * mu2;
            float g  = __fdividef(sGP, sGT + EPS_V);
            float sv = sP - g * sGP;
            if (sGT < EPS_V) { g = 0.f; sv = sP; sGT = 0.f; }
            if (sP  < EPS_V) { g = 0.f; sv = 0.f; }
            if (g < 0.f)     { sv = sP; g = 0.f; }
            if (sv <= EPS_V) sv = EPS_V;
            nsum += __log10f(1.f + __fdividef(g * g * sGT, sv + 2.0f));
            dsum += __log10f(1.f + sGT * 0.5f);
        }
    }
    for (int off = 16; off > 0; off >>= 1) {
        nsum += __shfl_xor(nsum, off);
        dsum += __shfl_xor(dsum, off);
    }
    if (lane == 0) {
        atomicAdd(&numAcc[img], nsum);
        atomicAdd(&denAcc[img], dsum);
    }
}

// ---------------------------------------------------------------------------
__global__ void zero_ws_kernel(float* __restrict__ p, int n)
{
    for (int i = blockIdx.x * blockDim.x + threadIdx.x; i < n;
         i += gridDim.x * blockDim.x)
        p[i] = 0.f;
}

// Per-image dot of (x - recons) with W  (the +b cancels in the difference).
__global__ __launch_bounds__(256)
void pred_kernel(const float* __restrict__ recons, const float* __restrict__ xin,
                 const float* __restrict__ Wlin, float* __restrict__ dots)
{
    __shared__ float red[256];
    const int b = blockIdx.x, t = threadIdx.x;
    const float4* r4 = (const float4*)(recons + (size_t)b * CHW);
    const float4* x4 = (const float4*)(xin + (size_t)b * CHW);
    const float4* w4 = (const float4*)Wlin;
    float acc = 0.f;
    for (int i = t; i < CHW / 4; i += 256) {
        const float4 rv = r4[i], xv = x4[i], wv = w4[i];
        acc += (xv.x - rv.x) * wv.x + (xv.y - rv.y) * wv.y
             + (xv.z - rv.z) * wv.z + (xv.w - rv.w) * wv.w;
    }
    red[t] = acc;
    __syncthreads();
    for (int s = 128; s > 0; s >>= 1) {
        if (t < s) red[t] += red[t + s];
        __syncthreads();
    }
    if (t == 0) dots[b] = red[0];
}

__global__ __launch_bounds__(128)
void finalize_kernel(const float* __restrict__ numAcc,
                     const float* __restrict__ denAcc,
                     const float* __restrict__ dots, float* __restrict__ out)
{
    __shared__ float s1[128], s2[128];
    const int t = threadIdx.x;
    float vif = 0.f;
    for (int c = 0; c < N_C; ++c)
        vif += numAcc[t * N_C + c] / denAcc[t * N_C + c];
    vif *= (1.0f / N_C);
    s1[t] = 1.0f - vif;
    const float d = dots[t];
    s2[t] = d * d;
    __syncthreads();
    for (int s = 64; s > 0; s >>= 1) {
        if (t < s) { s1[t] += s1[t + s]; s2[t] += s2[t + s]; }
        __syncthreads();
    }
    if (t == 0) {
        const float recons_loss = s1[0] * (1.0f / N_B);
        const float pred_loss   = s2[0] * (1.0f / N_B);
        out[0] = pred_loss + recons_loss;
        out[1] = recons_loss;
        out[2] = pred_loss;
    }
}

// ---------------------------------------------------------------------------
extern "C" void kernel_launch(void* const* d_in, const int* in_sizes, int n_in,
                              void* d_out, int out_size, void* d_ws, size_t ws_size,
                              hipStream_t stream)
{
    const float* recons = (const float*)d_in[0];
    const float* xin    = (const float*)d_in[1];
    const float* Wlin   = (const float*)d_in[2];
    float* out = (float*)d_out;

    float* wsf    = (float*)d_ws;
    float* numAcc = wsf;                 // N_IMG floats
    float* denAcc = wsf + N_IMG;         // N_IMG floats
    float* dots   = wsf + 2 * N_IMG;     // N_B floats

    zero_ws_kernel<<<1, 256, 0, stream>>>(wsf, 2 * N_IMG + N_B);

    #define LAUNCH_SCALE(NN)                                                   \
        {                                                                      \
            dim3 g(((IMG_W - (NN) + 1) + 15) / 16,                             \
                   ((IMG_H - (NN) + 1) + 15) / 16, N_IMG);                     \
            vif_scale_kernel<NN><<<g, 32, 0, stream>>>(recons, xin,            \
                                                       numAcc, denAcc);        \
        }
    LAUNCH_SCALE(17)
    LAUNCH_SCALE(9)
    LAUNCH_SCALE(5)
    LAUNCH_SCALE(3)
    #undef LAUNCH_SCALE

    pred_kernel<<<N_B, 256, 0, stream>>>(recons, xin, Wlin, dots);
    finalize_kernel<<<1, 128, 0, stream>>>(numAcc, denAcc, dots, out);
}